// gwasVAE_51479478009903
// MI455X (gfx1250) — compile-verified
//
#include <hip/hip_runtime.h>
#include <hip/hip_bf16.h>

// ---------- CDNA5 WMMA types ----------
typedef float v2f __attribute__((ext_vector_type(2)));
typedef float v8f __attribute__((ext_vector_type(8)));

#define B_DIM   64
#define H_DIM   2048
#define Z_DIM   64
#define TWO_N   8192
#define N_POP   4096.0f
#define SCALED_POP_MUT 2.048e-4f   // 4*N*U = 4*4096*1.25e-8

// ---------------------------------------------------------------------------
// Generic fp32 GEMM + bias via V_WMMA_F32_16X16X4_F32.
// One wave (32 lanes) computes one 16x16 tile of C = A[M,K] @ W[K,N] + bias.
// A fragment (16x4, ISA layout): lane l holds A[tm + (l&15), k0 + 2*(l>>4) + {0,1}]
// B fragment (4x16, dual):       lane l holds W[k0 + 2*(l>>4) + {0,1}, tn + (l&15)]
// C/D layout: VGPR v, lanes 0-15 -> row tm+v, lanes 16-31 -> row tm+v+8.
// ---------------------------------------------------------------------------
__global__ __launch_bounds__(32) void gemm_wmma_f32(
    const float* __restrict__ A, const float* __restrict__ W,
    const float* __restrict__ bias, float* __restrict__ C,
    int M, int K, int N)
{
    const int lane = threadIdx.x & 31;
    const int l    = lane & 15;
    const int half = lane >> 4;
    const int tn   = blockIdx.x * 16;
    const int tm   = blockIdx.y * 16;

    const float* __restrict__ aRow = A + (size_t)(tm + l) * (size_t)K;
    const float* __restrict__ wCol = W + (size_t)(tn + l);

    v8f acc = {};
    #pragma unroll 8
    for (int k0 = 0; k0 < K; k0 += 4) {
        const int ka = k0 + 2 * half;
        v2f a = *(const v2f*)(aRow + ka);                 // global_load_b64
        v2f b;
        b.x = wCol[(size_t)ka * (size_t)N];
        b.y = wCol[(size_t)(ka + 1) * (size_t)N];
        acc = __builtin_amdgcn_wmma_f32_16x16x4_f32(
                  /*neg_a=*/false, a, /*neg_b=*/false, b,
                  /*c_mod=*/(short)0, acc,
                  /*reuse_a=*/false, /*reuse_b=*/false);
    }

    const float bv = bias[tn + l];
    const int   m0 = tm + half * 8;
    #pragma unroll
    for (int v = 0; v < 8; ++v)
        C[(size_t)(m0 + v) * (size_t)N + (size_t)(tn + l)] = acc[v] + bv;
}

// ---------------------------------------------------------------------------
// Heads: softplus -> clip(0.01,7) for mu/var, z = eps*sqrt(var)+mu,
// write z_mu / z_var / z (twice) into d_out, block-partial sums of z.
// ---------------------------------------------------------------------------
__device__ __forceinline__ float softplus_clip(float x) {
    float sp = fmaxf(x, 0.0f) + log1pf(expf(-fabsf(x)));
    return fminf(fmaxf(sp, 0.01f), 7.0f);
}

__global__ __launch_bounds__(256) void heads_kernel(
    const float* __restrict__ Hm, const float* __restrict__ Hv,
    const float* __restrict__ eps, float* __restrict__ out,
    float* __restrict__ partials)
{
    const int t  = blockIdx.x * 256 + threadIdx.x;     // 0..4095
    const float mu  = softplus_clip(Hm[t]);
    const float var = softplus_clip(Hv[t]);
    const float z   = fmaf(eps[t], sqrtf(var), mu);

    out[TWO_N + t]                 = mu;   // z_mu   [8192,12288)
    out[TWO_N + 4096 + t]          = var;  // z_var  [12288,16384)
    out[TWO_N + 8192 + 1 + t]      = z;    // z      [16385,20481)
    out[TWO_N + 12288 + 2 + t]     = z;    // z      [20482,24578)

    __shared__ float red[256];
    red[threadIdx.x] = z;
    __syncthreads();
    #pragma unroll
    for (int s = 128; s > 0; s >>= 1) {
        if (threadIdx.x < s) red[threadIdx.x] += red[threadIdx.x + s];
        __syncthreads();
    }
    if (threadIdx.x == 0) partials[blockIdx.x] = red[0];
}

// ---------------------------------------------------------------------------
// Finalize s = mean(z); build decode lookup tables:
//   lp[j] = log(p_j), l1p[j] = log1p(-p_j),
//   num1w[j] = trapezoid_weight_j * (1 - exp(-2*N*s*(1-p_j)))
// Also denom_inv and the fixed outputs x_mean[0]=0, log_det_j=0.
// ---------------------------------------------------------------------------
__global__ __launch_bounds__(256) void tables_kernel(
    const float* __restrict__ partials, float* __restrict__ out,
    float* __restrict__ scal, float* __restrict__ lp,
    float* __restrict__ l1p, float* __restrict__ num1w)
{
    __shared__ float sh_twoNs;
    if (threadIdx.x == 0) {
        float s = 0.0f;
        for (int i = 0; i < 16; ++i) s += partials[i];
        s *= (1.0f / 4096.0f);
        const float twoNs = 2.0f * N_POP * s;
        scal[0] = s;
        scal[1] = 1.0f / (1.0f - expf(-twoNs));   // denom_inv
        sh_twoNs = twoNs;
        out[0]            = 0.0f;                 // x_mean[0]
        out[TWO_N + 8192] = 0.0f;                 // log_det_j @ 16384
    }
    __syncthreads();
    const float twoNs = sh_twoNs;
    const float dxp   = 0.999f / 8191.0f;
    for (int j = threadIdx.x; j < TWO_N; j += 256) {
        float p = fmaf((float)j, dxp, 0.001f);
        if (j == TWO_N - 1) p = 1.0f;
        lp[j]  = logf(p);
        l1p[j] = log1pf(-p);
        const float w = (j == 0 || j == TWO_N - 1) ? 0.5f * dxp : dxp;
        num1w[j] = w * (1.0f - expf(-twoNs * (1.0f - p)));
    }
}

// ---------------------------------------------------------------------------
// Decode: one block per i in 1..2N-1.
//   x_mean[i] = 4*N*U * denom_inv * C(i) * sum_j num1w[j]*exp((i-1)lp + (N-i-1)l1p)
// ---------------------------------------------------------------------------
__global__ __launch_bounds__(256) void decode_kernel(
    const float* __restrict__ binom, const float* __restrict__ scal,
    const float* __restrict__ lp, const float* __restrict__ l1p,
    const float* __restrict__ num1w, float* __restrict__ out)
{
    const int   i  = blockIdx.x + 1;
    const float af = (float)(i - 1);
    const float bf = (float)(4095 - i);            // N - i - 1

    float acc = 0.0f;
    #pragma unroll 4
    for (int j = threadIdx.x; j < TWO_N; j += 256)
        acc += num1w[j] * __expf(fmaf(af, lp[j], bf * l1p[j]));

    __shared__ float red[256];
    red[threadIdx.x] = acc;
    __syncthreads();
    #pragma unroll
    for (int s = 128; s > 0; s >>= 1) {
        if (threadIdx.x < s) red[threadIdx.x] += red[threadIdx.x + s];
        __syncthreads();
    }
    if (threadIdx.x == 0)
        out[i] = SCALED_POP_MUT * scal[1] * binom[i] * red[0];
}

// ---------------------------------------------------------------------------
extern "C" void kernel_launch(void* const* d_in, const int* in_sizes, int n_in,
                              void* d_out, int out_size, void* d_ws, size_t ws_size,
                              hipStream_t stream) {
    const float* x   = (const float*)d_in[0];
    const float* W1  = (const float*)d_in[1];
    const float* b1  = (const float*)d_in[2];
    const float* W2  = (const float*)d_in[3];
    const float* b2  = (const float*)d_in[4];
    const float* W3  = (const float*)d_in[5];
    const float* b3  = (const float*)d_in[6];
    const float* Wm  = (const float*)d_in[7];
    const float* bm  = (const float*)d_in[8];
    const float* Wv  = (const float*)d_in[9];
    const float* bv  = (const float*)d_in[10];
    const float* eps = (const float*)d_in[11];
    const float* bc  = (const float*)d_in[12];
    float* out = (float*)d_out;

    float* ws       = (float*)d_ws;
    float* h1       = ws;                       // 64*2048
    float* h2       = ws + 131072;              // 64*2048
    float* h3       = ws + 262144;              // 64*2048
    float* Hm       = ws + 393216;              // 64*64
    float* Hv       = ws + 397312;              // 64*64
    float* partials = ws + 401408;              // 16
    float* scal     = ws + 401472;              // s, denom_inv
    float* lp       = ws + 401536;              // 8192
    float* l1p      = lp + TWO_N;               // 8192
    float* num1w    = l1p + TWO_N;              // 8192

    const dim3 wave(32);
    // encoder: h = ((x@W1+b1)@W2+b2)@W3+b3
    gemm_wmma_f32<<<dim3(H_DIM / 16, B_DIM / 16), wave, 0, stream>>>(x,  W1, b1, h1, B_DIM, TWO_N, H_DIM);
    gemm_wmma_f32<<<dim3(H_DIM / 16, B_DIM / 16), wave, 0, stream>>>(h1, W2, b2, h2, B_DIM, H_DIM, H_DIM);
    gemm_wmma_f32<<<dim3(H_DIM / 16, B_DIM / 16), wave, 0, stream>>>(h2, W3, b3, h3, B_DIM, H_DIM, H_DIM);
    // heads (pre-activation)
    gemm_wmma_f32<<<dim3(Z_DIM / 16, B_DIM / 16), wave, 0, stream>>>(h3, Wm, bm, Hm, B_DIM, H_DIM, Z_DIM);
    gemm_wmma_f32<<<dim3(Z_DIM / 16, B_DIM / 16), wave, 0, stream>>>(h3, Wv, bv, Hv, B_DIM, H_DIM, Z_DIM);
    // activations + reparameterize + partial sums for mean(z)
    heads_kernel<<<16, 256, 0, stream>>>(Hm, Hv, eps, out, partials);
    // scalar mean, denom_inv, decode tables, fixed outputs
    tables_kernel<<<1, 256, 0, stream>>>(partials, out, scal, lp, l1p, num1w);
    // 8191 rows of the integral decoder
    decode_kernel<<<TWO_N - 1, 256, 0, stream>>>(bc, scal, lp, l1p, num1w, out);
}